// Mamba_57621281243605
// MI455X (gfx1250) — compile-verified
//
#include <hip/hip_runtime.h>
#include <hip/hip_bf16.h>
#include <math.h>

typedef __attribute__((ext_vector_type(2))) float v2f;
typedef __attribute__((ext_vector_type(8))) float v8f;

#define HW    576          // 24*24
#define NTOK  1152         // N(2) * HW
#define DM    96
#define DI    192
#define KP    866          // padded LDS row stride (dwords): even (b64-aligned), bank-spread

__device__ __forceinline__ float sigmoidf_(float x) { return 1.0f / (1.0f + expf(-x)); }
__device__ __forceinline__ float siluf_(float x)    { return x * sigmoidf_(x); }

// ---------------- rmsnorm over W axis (24 elems), wave per (n,c,h) row ----
__global__ void k_rmsnorm(const float* __restrict__ X, long cs, long ns,
                          const float* __restrict__ nw, float* __restrict__ xn)
{
    int wid  = blockIdx.x * (blockDim.x >> 5) + (threadIdx.x >> 5);
    int lane = threadIdx.x & 31;
    if (wid >= 2 * DM * 24) return;
    int h = wid % 24;
    int c = (wid / 24) % DM;
    int n = wid / (24 * DM);
    const float* src = X + (long)n * ns + (long)c * cs + h * 24;
    float v = (lane < 24) ? src[lane] : 0.0f;
    float s = v * v;
    #pragma unroll
    for (int off = 16; off >= 1; off >>= 1) s += __shfl_xor(s, off, 32);
    float scale = rsqrtf(s * (1.0f / 24.0f) + 1e-5f) * nw[c];
    if (lane < 24) xn[(n * DM + c) * HW + h * 24 + lane] = v * scale;
}

// ---------------- in_proj 1x3x3 conv as implicit GEMM (M=384,K=864,N=16/blk)
// LDS-staged im2col panel; 8 waves x 3 M-tiles; fused conv1d-tap3+SiLU (xi)
// and SiLU (z) epilogues, outputs token-major [t][192].
__global__ __launch_bounds__(256) void
k_inproj(const float* __restrict__ Wp, const float* __restrict__ xn,
         const float* __restrict__ cw, const float* __restrict__ cb,
         float* __restrict__ xi_t, float* __restrict__ zs_t)
{
    __shared__ float lds[16 * KP];
    const int tid = threadIdx.x;
    const int t0  = blockIdx.x * 16;          // token tile (72 blocks)

    // ---- stage phase: lds[tok*KP + k] = im2col B[k, t0+tok]
    {
        int tok = tid & 15;
        int t = t0 + tok;
        int n = t / HW, hw = t % HW;
        int h = hw / 24, w = hw % 24;
        int ks  = tid >> 4;                   // 0..15
        int ic  = ks / 9, pos = ks % 9;       // incremental thereafter
        for (int k = ks; k < 864; k += 16) {
            int hh = h + pos / 3 - 1;
            int ww = w + pos % 3 - 1;
            float val = 0.0f;
            if ((unsigned)hh < 24u && (unsigned)ww < 24u)
                val = xn[(n * DM + ic) * HW + hh * 24 + ww];
            lds[tok * KP + k] = val;
            pos += 16;
            while (pos >= 9) { pos -= 9; ++ic; }
        }
    }
    __syncthreads();

    // ---- compute phase: no divergence, EXEC all-ones around WMMA
    const int wave = tid >> 5, lane = tid & 31;
    const int lo = lane & 15, hi = lane >> 4;
    const int mb = wave * 48;                 // 3 M-tiles per wave
    const v2f* A0 = (const v2f*)(Wp + (long)(mb +      lo) * 864);
    const v2f* A1 = (const v2f*)(Wp + (long)(mb + 16 + lo) * 864);
    const v2f* A2 = (const v2f*)(Wp + (long)(mb + 32 + lo) * 864);
    const float* Brow = lds + lo * KP + 2 * hi;
    v8f acc0 = {}, acc1 = {}, acc2 = {};
    for (int k0 = 0; k0 < 864; k0 += 4) {
        int ki = (k0 >> 1) + hi;              // v2f index -> floats k0+2*hi, +1
        v2f b  = *(const v2f*)(Brow + k0);
        acc0 = __builtin_amdgcn_wmma_f32_16x16x4_f32(false, A0[ki], false, b, (short)0, acc0, false, false);
        acc1 = __builtin_amdgcn_wmma_f32_16x16x4_f32(false, A1[ki], false, b, (short)0, acc1, false, false);
        acc2 = __builtin_amdgcn_wmma_f32_16x16x4_f32(false, A2[ki], false, b, (short)0, acc2, false, false);
    }

    // ---- epilogue: rows<192 -> xi = silu(conv1d_b + w3*v); rows>=192 -> silu(z)
    const int t = t0 + lo;
    #pragma unroll
    for (int q = 0; q < 3; ++q) {
        v8f acc = (q == 0) ? acc0 : (q == 1) ? acc1 : acc2;
        int base = mb + q * 16;               // multiple of 16: branch is uniform
        if (base < DI) {
            #pragma unroll
            for (int j = 0; j < 8; ++j) {
                int row = base + j + 8 * hi;
                float pre = cb[row] + cw[row * 4 + 3] * acc[j];
                xi_t[(long)t * DI + row] = siluf_(pre);
            }
        } else {
            #pragma unroll
            for (int j = 0; j < 8; ++j) {
                int row = base - DI + j + 8 * hi;
                zs_t[(long)t * DI + row] = siluf_(acc[j]);
            }
        }
    }
}

// ---------------- x_proj: M=48(pad of 38) K=192 N=1152, B contiguous ------
__global__ void k_xproj(const float* __restrict__ xpw, const float* __restrict__ xi_t,
                        float* __restrict__ dbc)
{
    int wid  = blockIdx.x * (blockDim.x >> 5) + (threadIdx.x >> 5); // 0..215
    int lane = threadIdx.x & 31;
    int lo = lane & 15, hi = lane >> 4;
    int mt = wid % 3, tt = wid / 3;
    int m0 = mt * 16, t0 = tt * 16;
    int m  = m0 + lo;
    float msk = (m < 38) ? 1.0f : 0.0f;
    const v2f* Av = (const v2f*)(xpw + (long)((m < 38) ? m : 0) * DI);
    const float* Brow = xi_t + (long)(t0 + lo) * DI + 2 * hi;
    v8f acc = {};
    for (int k0 = 0; k0 < DI; k0 += 4) {
        v2f a = Av[(k0 >> 1) + hi];
        a.x *= msk; a.y *= msk;
        v2f b = *(const v2f*)(Brow + k0);
        acc = __builtin_amdgcn_wmma_f32_16x16x4_f32(false, a, false, b, (short)0, acc, false, false);
    }
    #pragma unroll
    for (int j = 0; j < 8; ++j) {
        int row = m0 + j + 8 * hi;            // dbc buffer has 48 rows
        dbc[(long)row * NTOK + (t0 + lo)] = acc[j];
    }
}

// ---------------- BC[t] = sum_s B[s]*C[s]  (l=0: hs == BX) ----------------
__global__ void k_bc(const float* __restrict__ dbc, float* __restrict__ BC)
{
    int t = blockIdx.x * blockDim.x + threadIdx.x;
    if (t >= NTOK) return;
    float s = 0.0f;
    #pragma unroll
    for (int i = 0; i < 16; ++i)
        s += dbc[(6 + i) * NTOK + t] * dbc[(22 + i) * NTOK + t];
    BC[t] = s;
}

// ---------------- inner = (softplus(dt)*xi*BC + D*xi) * silu(z), token-major
__global__ void k_inner(const float* __restrict__ dbc, const float* __restrict__ dtw,
                        const float* __restrict__ dtb, const float* __restrict__ xi_t,
                        const float* __restrict__ Dp, const float* __restrict__ BC,
                        const float* __restrict__ zs_t, float* __restrict__ inner_t)
{
    int i = blockIdx.x * blockDim.x + threadIdx.x;
    if (i >= DI * NTOK) return;
    int t = i / DI, c = i - t * DI;
    float dr = dtb[c];
    #pragma unroll
    for (int r = 0; r < 6; ++r) dr += dtw[c * 6 + r] * dbc[r * NTOK + t];
    float delta = (dr > 20.0f) ? dr : log1pf(expf(dr));
    float xiv = xi_t[i];
    float y   = delta * xiv * BC[t] + Dp[c] * xiv;
    inner_t[i] = y * zs_t[i];
}

// ---------------- out_proj + residual (residual preloaded into C) ---------
__global__ void k_outproj(const float* __restrict__ opw, const float* __restrict__ inner_t,
                          const float* __restrict__ Res, long rcs, long rns,
                          float* __restrict__ Out)
{
    int wid  = blockIdx.x * (blockDim.x >> 5) + (threadIdx.x >> 5); // 0..431
    int lane = threadIdx.x & 31;
    int lo = lane & 15, hi = lane >> 4;
    int mt = wid % 6, tt = wid / 6;
    int m0 = mt * 16, t0 = tt * 16;
    int t  = t0 + lo;
    int n  = t / HW, hw = t % HW;
    v8f acc;
    #pragma unroll
    for (int j = 0; j < 8; ++j) {
        int row = m0 + j + 8 * hi;
        acc[j] = Res[(long)n * rns + (long)row * rcs + hw];
    }
    const v2f* Av = (const v2f*)(opw + (long)(m0 + lo) * DI);
    const float* Brow = inner_t + (long)t * DI + 2 * hi;
    for (int k0 = 0; k0 < DI; k0 += 4) {
        v2f a = Av[(k0 >> 1) + hi];
        v2f b = *(const v2f*)(Brow + k0);
        acc = __builtin_amdgcn_wmma_f32_16x16x4_f32(false, a, false, b, (short)0, acc, false, false);
    }
    #pragma unroll
    for (int j = 0; j < 8; ++j) {
        int row = m0 + j + 8 * hi;
        Out[(long)n * (DM * HW) + (long)row * HW + hw] = acc[j];
    }
}

extern "C" void kernel_launch(void* const* d_in, const int* in_sizes, int n_in,
                              void* d_out, int out_size, void* d_ws, size_t ws_size,
                              hipStream_t stream)
{
    const float* x    = (const float*)d_in[0];
    const float* inw  = (const float*)d_in[1];   // (2,384,96,1,3,3)
    const float* cw   = (const float*)d_in[2];   // (2,192,1,4,1,1)
    const float* cb   = (const float*)d_in[3];   // (2,192)
    const float* xpw  = (const float*)d_in[4];   // (2,38,192)
    const float* dtw  = (const float*)d_in[5];   // (2,192,6)
    const float* dtb  = (const float*)d_in[6];   // (2,192)
    /* d_in[7] = A_log: provably unused at l=0 (inclusive scan: hs[0]=BX[0]) */
    const float* Dp   = (const float*)d_in[8];   // (2,192)
    const float* opw  = (const float*)d_in[9];   // (2,96,192)
    const float* nw   = (const float*)d_in[10];  // (2,1,96,1,1,1)
    float* out = (float*)d_out;
    float* ws  = (float*)d_ws;

    float* xn      = ws;                  // 110592
    float* xi_t    = xn      + 110592;    // 221184  [t][192]
    float* zs_t    = xi_t    + 221184;    // 221184  [t][192] (silu applied)
    float* dbc     = zs_t    + 221184;    // 55296   (48 rows x NTOK, 38 used)
    float* BC      = dbc     + 55296;     // 1152
    float* inner_t = BC      + 1152;      // 221184  [t][192]
    float* y0      = inner_t + 221184;    // 110592

    for (int l = 0; l < 2; ++l) {
        const float* Xin; long cs, ns;
        if (l == 0) { Xin = x;  cs = 8 * HW; ns = (long)DM * 8 * HW; }
        else        { Xin = y0; cs = HW;     ns = (long)DM * HW;     }
        float* Out = (l == 0) ? y0 : out;

        k_rmsnorm<<<576, 256, 0, stream>>>(Xin, cs, ns, nw + l * DM, xn);
        k_inproj <<<72, 256, 0, stream>>>(inw + (long)l * 384 * 864, xn,
                                          cw + l * DI * 4, cb + l * DI, xi_t, zs_t);
        k_xproj  <<<54, 128, 0, stream>>>(xpw + l * 38 * DI, xi_t, dbc);
        k_bc     <<<(NTOK + 255) / 256, 256, 0, stream>>>(dbc, BC);
        k_inner  <<<(DI * NTOK + 255) / 256, 256, 0, stream>>>(dbc, dtw + l * DI * 6, dtb + l * DI,
                                                               xi_t, Dp + l * DI, BC, zs_t, inner_t);
        k_outproj<<<108, 128, 0, stream>>>(opw + l * DM * DI, inner_t, Xin, cs, ns, Out);
    }
}